// enhance_74131135529025
// MI455X (gfx1250) — compile-verified
//
#include <hip/hip_runtime.h>
#include <hip/hip_bf16.h>

typedef __attribute__((ext_vector_type(2))) float v2f;
typedef __attribute__((ext_vector_type(8))) float v8f;

#define B_    16
#define C_    64
#define HW_   65536          // 256*256
#define NBINS 256
#define EPSN  1e-12f

// ---------------- Stage A: per-(b,c) plane mean -> a[1024] ----------------
__global__ void mean_kernel(const float* __restrict__ F, float* __restrict__ a) {
    int plane = blockIdx.x;                       // 0..1023 = b*64+c
    const float4* src = (const float4*)(F + (size_t)plane * HW_);
    float s = 0.f;
    for (int i = threadIdx.x; i < HW_ / 4; i += 256) {
        float4 v = src[i];
        s += v.x + v.y + v.z + v.w;
    }
    for (int off = 16; off; off >>= 1) s += __shfl_down(s, off);
    __shared__ float red[8];
    if ((threadIdx.x & 31) == 0) red[threadIdx.x >> 5] = s;
    __syncthreads();
    if (threadIdx.x == 0) {
        float t = 0.f;
        for (int i = 0; i < 8; i++) t += red[i];
        a[plane] = t * (1.0f / (float)HW_);
    }
}

// ---------------- Stage B: L2-normalize a over channels -> an[1024] -------
__global__ void norm_kernel(const float* __restrict__ a, float* __restrict__ an) {
    int b = blockIdx.x;                           // 16 blocks x 64 threads
    int c = threadIdx.x;
    float v  = a[b * C_ + c];
    float sq = v * v;
    for (int off = 16; off; off >>= 1) sq += __shfl_down(sq, off);
    __shared__ float red[2];
    if ((threadIdx.x & 31) == 0) red[threadIdx.x >> 5] = sq;
    __syncthreads();
    float n = sqrtf(red[0] + red[1]);
    an[b * C_ + c] = v / fmaxf(n, EPSN);
}

// ---------------- zero the per-batch histograms ---------------------------
__global__ void zero_hist_kernel(unsigned int* __restrict__ hist) {
    hist[blockIdx.x * NBINS + threadIdx.x] = 0u;
}

// ------- Stage C: cos-sim via WMMA f32 16x16x4, quantize, histogram -------
// grid = B_*32 blocks of 256 threads (8 waves); each block handles 2048 pixels.
#define PIXPB 2048
#define CHUNKS (HW_ / PIXPB)     // 32
__global__ void cos_hist_kernel(const float* __restrict__ F,
                                const float* __restrict__ an,
                                unsigned char* __restrict__ qout,
                                unsigned int* __restrict__ hist) {
    const int b       = blockIdx.x / CHUNKS;
    const int pixBase = (blockIdx.x % CHUNKS) * PIXPB;
    const int wave    = threadIdx.x >> 5;
    const int lane    = threadIdx.x & 31;
    const int half    = lane >> 4;                 // 0: K=0,1  1: K=2,3
    const int nidx    = lane & 15;                 // pixel column within group

    __shared__ float        s_an[C_];
    __shared__ unsigned int s_hist[NBINS];
    if (threadIdx.x < C_)  s_an[threadIdx.x]  = an[b * C_ + threadIdx.x];
    s_hist[threadIdx.x] = 0u;
    __syncthreads();

    const float* Fb = F + ((size_t)b << 22);       // b * C_ * HW_

    for (int it = 0; it < PIXPB / 128; ++it) {     // 8 waves * 16 pixels = 128
        const int p = pixBase + it * 128 + wave * 16 + nidx;

        v8f  acc = {};
        float ss = 0.f;
#pragma unroll
        for (int k0 = 0; k0 < C_; k0 += 4) {
            const int c0 = k0 + half * 2;
            const int c1 = c0 + 1;
            float f0 = Fb[((size_t)c0 << 16) + p];
            float f1 = Fb[((size_t)c1 << 16) + p];
            ss = fmaf(f0, f0, ss);
            ss = fmaf(f1, f1, ss);
            v2f Am = { s_an[c0], s_an[c1] };       // A rows all identical
            v2f Bm = { f0, f1 };                   // B[k][n] for this lane
            acc = __builtin_amdgcn_wmma_f32_16x16x4_f32(
                      false, Am, false, Bm, (short)0, acc, false, false);
        }
        // combine ||F||^2 halves across the two 16-lane groups
        ss += __shfl_xor(ss, 16);
        float dot  = acc[0];                       // D[0, n] == D[8, n] == dot_n
        float cosv = dot / fmaxf(sqrtf(ss), EPSN);
        int   qv   = ((int)truncf(cosv * 255.0f)) & 255;
        if (lane < 16) {
            qout[(size_t)b * HW_ + p] = (unsigned char)qv;
            atomicAdd(&s_hist[qv], 1u);
        }
    }
    __syncthreads();
    atomicAdd(&hist[b * NBINS + threadIdx.x], s_hist[threadIdx.x]);
}

// ------------- Stage D: per-batch CDF -> equalization LUT (/255) ----------
__global__ void lut_kernel(const unsigned int* __restrict__ hist,
                           float* __restrict__ lut) {
    const int b = blockIdx.x;
    const int t = threadIdx.x;                     // 256 threads
    __shared__ float cdf[NBINS];
    __shared__ float red[NBINS];
    float hv = (float)hist[b * NBINS + t];
    cdf[t] = hv;
    __syncthreads();
    for (int off = 1; off < NBINS; off <<= 1) {    // inclusive scan
        float add = (t >= off) ? cdf[t - off] : 0.0f;
        __syncthreads();
        cdf[t] += add;
        __syncthreads();
    }
    float c = cdf[t];
    red[t] = (hv > 0.0f) ? c : ((float)HW_ + 1.0f);
    __syncthreads();
    for (int off = NBINS / 2; off; off >>= 1) {    // min-reduce -> cdf_min
        if (t < off) red[t] = fminf(red[t], red[t + off]);
        __syncthreads();
    }
    float cdfmin = red[0];
    float denom  = fmaxf((float)HW_ - cdfmin, 1.0f);
    float l = rintf((c - cdfmin) * (255.0f / denom));   // round-half-even
    l = fminf(fmaxf(l, 0.0f), 255.0f);
    lut[b * NBINS + t] = l * (1.0f / 255.0f);
}

// --------------- Stage E: out = lut[q] * F (streaming) --------------------
__global__ void apply_kernel(const float* __restrict__ F,
                             const unsigned char* __restrict__ q,
                             const float* __restrict__ lut,
                             float* __restrict__ out) {
    const int plane = blockIdx.x;                  // b*64+c
    const int b     = plane >> 6;
    __shared__ float slut[NBINS];
    slut[threadIdx.x] = lut[b * NBINS + threadIdx.x];
    __syncthreads();
    const float4* src = (const float4*)(F   + (size_t)plane * HW_);
    float4*       dst = (float4*)      (out + (size_t)plane * HW_);
    const uchar4* qq  = (const uchar4*)(q   + (size_t)b * HW_);
    for (int i = threadIdx.x; i < HW_ / 4; i += 256) {
        float4 v = src[i];
        uchar4 u = qq[i];
        float4 r;
        r.x = v.x * slut[u.x];
        r.y = v.y * slut[u.y];
        r.z = v.z * slut[u.z];
        r.w = v.w * slut[u.w];
        dst[i] = r;
    }
}

extern "C" void kernel_launch(void* const* d_in, const int* in_sizes, int n_in,
                              void* d_out, int out_size, void* d_ws, size_t ws_size,
                              hipStream_t stream) {
    (void)in_sizes; (void)n_in; (void)out_size; (void)ws_size;
    const float* F = (const float*)d_in[0];
    float* out = (float*)d_out;

    char* ws = (char*)d_ws;
    float*         a    = (float*)(ws + 0);                 //  4 KB
    float*         an   = (float*)(ws + 4096);              //  4 KB
    unsigned int*  hist = (unsigned int*)(ws + 8192);       // 16 KB
    float*         lut  = (float*)(ws + 8192 + 16384);      // 16 KB
    unsigned char* q    = (unsigned char*)(ws + 40960);     //  1 MB

    mean_kernel     <<<B_ * C_,      256, 0, stream>>>(F, a);
    norm_kernel     <<<B_,            64, 0, stream>>>(a, an);
    zero_hist_kernel<<<B_,           256, 0, stream>>>(hist);
    cos_hist_kernel <<<B_ * CHUNKS,  256, 0, stream>>>(F, an, q, hist);
    lut_kernel      <<<B_,           256, 0, stream>>>(hist, lut);
    apply_kernel    <<<B_ * C_,      256, 0, stream>>>(F, q, lut, out);
}